// GIN_20529943675473
// MI455X (gfx1250) — compile-verified
//
#include <hip/hip_runtime.h>

// GIN 2-layer forward for MI455X (gfx1250, wave32).
// fp32 end-to-end to match the reference; GEMMs use V_WMMA_F32_16X16X4_F32.

#define D 128                 // feature dim
#define TILE_ROWS 16          // rows of nodes per workgroup
#define BLOCK 256             // 8 wave32 waves per workgroup

typedef float v2f __attribute__((ext_vector_type(2)));
typedef float v8f __attribute__((ext_vector_type(8)));

// ---------------------------------------------------------------------------
// Edge scatter: agg[dst[e]][c] += x[src[e]][c]; one thread per (edge, channel).
// agg (51.2 MB) is L2-resident -> global_atomic_add_f32 resolves in L2.
// ---------------------------------------------------------------------------
__global__ __launch_bounds__(BLOCK)
void gin_scatter_add(const float* __restrict__ x,
                     const int*   __restrict__ src,
                     const int*   __restrict__ dst,
                     float*       __restrict__ agg,
                     int nEdges) {
    long gid = (long)blockIdx.x * blockDim.x + threadIdx.x;
    long total = (long)nEdges * D;
    if (gid >= total) return;
    int e = (int)(gid >> 7);          // D == 128
    int c = (int)(gid & (D - 1));
    int s = src[e];
    int d = dst[e];
    float v = x[(long)s * D + c];
    unsafeAtomicAdd(&agg[(long)d * D + c], v);   // -> global_atomic_add_f32
}

// ---------------------------------------------------------------------------
// Fused GIN MLP on a 16-row node tile:
//   s = x + agg            (LDS)
//   t = relu(s @ W1 + b1)  (LDS)
//   u = t @ W2 + b2        -> out (optionally relu'd)
// Each of the 8 waves owns one 16-col output tile; K=128 in 32 WMMA steps.
//
// f32 WMMA fragment layouts (ISA 7.12.2):
//   A 16x4 : lane L<16 holds A[M=L][K=k+0], A[M=L][K=k+1] in v0,v1
//            lane L>=16 holds A[M=L-16][K=k+2], A[M=L-16][K=k+3]
//   B 4x16 : lane L<16 holds B[K=k+0][N=L], B[K=k+1][N=L]
//            lane L>=16 holds B[K=k+2][N=L-16], B[K=k+3][N=L-16]
//   C/D    : VGPR v, lane L -> D[M = v + 8*(L>=16)][N = L&15]
// ---------------------------------------------------------------------------
__global__ __launch_bounds__(BLOCK)
void gin_mlp(const float* __restrict__ xin,
             const float* __restrict__ agg,
             const float* __restrict__ W1, const float* __restrict__ b1,
             const float* __restrict__ W2, const float* __restrict__ b2,
             float* __restrict__ out,
             int applyReluOut) {
    __shared__ float s_tile[TILE_ROWS][D];   // 8 KB
    __shared__ float t_tile[TILE_ROWS][D];   // 8 KB

    const int tid  = threadIdx.x;
    const int lane = tid & 31;
    const int wave = tid >> 5;               // 0..7 -> output col tile
    const long row0 = (long)blockIdx.x * TILE_ROWS;

    // Cooperative load: s = x + agg  (2048 floats / 256 threads = 8 each)
    #pragma unroll
    for (int i = tid; i < TILE_ROWS * D; i += BLOCK) {
        int r = i >> 7;
        int c = i & (D - 1);
        long g = (row0 + r) * D + c;
        s_tile[r][c] = xin[g] + agg[g];
    }
    __syncthreads();

    const int mrow  = lane & 15;             // A-fragment M row / B-fragment N col
    const int kHalf = (lane >> 4) << 1;      // 0 or 2: which K pair this lane holds
    const int n0    = wave * 16;
    const int col   = n0 + mrow;
    const int rbase = (lane >> 4) << 3;      // C/D row base: 0 or 8

    // ---- Stage 1: t = relu(s @ W1 + b1) ----
    {
        v8f acc = {};
        #pragma unroll 4
        for (int k = 0; k < D; k += 4) {
            const int kb = k + kHalf;
            v2f a, b;
            a[0] = s_tile[mrow][kb];
            a[1] = s_tile[mrow][kb + 1];
            b[0] = W1[(long)kb * D + col];
            b[1] = W1[(long)(kb + 1) * D + col];
            acc = __builtin_amdgcn_wmma_f32_16x16x4_f32(
                false, a, false, b, (short)0, acc, false, false);
        }
        const float bias = b1[col];
        #pragma unroll
        for (int v = 0; v < 8; ++v) {
            float t = acc[v] + bias;
            t_tile[rbase + v][col] = t > 0.0f ? t : 0.0f;
        }
    }
    __syncthreads();

    // ---- Stage 2: u = t @ W2 + b2 ; out = relu?(u) ----
    {
        v8f acc = {};
        #pragma unroll 4
        for (int k = 0; k < D; k += 4) {
            const int kb = k + kHalf;
            v2f a, b;
            a[0] = t_tile[mrow][kb];
            a[1] = t_tile[mrow][kb + 1];
            b[0] = W2[(long)kb * D + col];
            b[1] = W2[(long)(kb + 1) * D + col];
            acc = __builtin_amdgcn_wmma_f32_16x16x4_f32(
                false, a, false, b, (short)0, acc, false, false);
        }
        const float bias = b2[col];
        #pragma unroll
        for (int v = 0; v < 8; ++v) {
            float u = acc[v] + bias;
            if (applyReluOut) u = u > 0.0f ? u : 0.0f;
            out[(row0 + rbase + v) * D + col] = u;
        }
    }
}

// ---------------------------------------------------------------------------
// Host launcher: memset agg -> scatter -> MLP(relu) -> memset -> scatter -> MLP
// All work on `stream`; hipMemsetAsync is graph-capturable.
// Inputs (setup_inputs order): x, edge_index, W1, b1, W2, b2.
// ---------------------------------------------------------------------------
extern "C" void kernel_launch(void* const* d_in, const int* in_sizes, int n_in,
                              void* d_out, int out_size, void* d_ws, size_t ws_size,
                              hipStream_t stream) {
    const float* x  = (const float*)d_in[0];
    const int*   ei = (const int*)  d_in[1];
    const float* W1 = (const float*)d_in[2];
    const float* b1 = (const float*)d_in[3];
    const float* W2 = (const float*)d_in[4];
    const float* b2 = (const float*)d_in[5];
    float* out = (float*)d_out;

    const int N = in_sizes[0] / D;     // 100000
    const int E = in_sizes[1] / 2;     // 600000
    const int* src = ei;               // edge_index row 0
    const int* dst = ei + E;           // edge_index row 1

    // Workspace: agg[N][D] and h1[N][D]  (2 * 51.2 MB)
    float* agg = (float*)d_ws;
    float* h1  = agg + (size_t)N * D;
    const size_t aggBytes = (size_t)N * D * sizeof(float);

    const long scatterThreads = (long)E * D;
    const int  sBlocks = (int)((scatterThreads + BLOCK - 1) / BLOCK);
    const int  mBlocks = N / TILE_ROWS;   // 100000 % 16 == 0

    // Layer 1: h1 = relu(MLP(x + segsum(x)))
    hipMemsetAsync(agg, 0, aggBytes, stream);
    gin_scatter_add<<<sBlocks, BLOCK, 0, stream>>>(x, src, dst, agg, E);
    gin_mlp<<<mBlocks, BLOCK, 0, stream>>>(x, agg, W1, b1, W2, b2, h1, 1);

    // Layer 2: out = MLP(h1 + segsum(h1))
    hipMemsetAsync(agg, 0, aggBytes, stream);
    gin_scatter_add<<<sBlocks, BLOCK, 0, stream>>>(h1, src, dst, agg, E);
    gin_mlp<<<mBlocks, BLOCK, 0, stream>>>(h1, agg, W1, b1, W2, b2, out, 0);
}